// ConvLayer_13391708030008
// MI455X (gfx1250) — compile-verified
//
#include <hip/hip_runtime.h>
#include <math.h>

// CGCNN ConvLayer forward on MI455X (gfx1250), fp32 WMMA path.
// Pipeline: k0 zero stats -> k1 persistent WMMA GEMM (+col stats, materialize
// gated) -> k3 BN1 + sigmoid*softplus + sum over M (+BN2 stats) -> k4 skip
// WMMA GEMM -> k5 BN2 + softplus epilogue.

#define NATOM 100000
#define MNBR  12
#define AF    128
#define EF    64
#define KDIM  320            // 2A + E
#define CDIM  256            // 2A
#define NM    (NATOM * MNBR) // 1,200,000 rows of the big GEMM

#define LDA   324            // padded LDS row stride (floats): conflict-free b64 frag loads
#define LDW   324
#define LDW2  132

typedef __attribute__((ext_vector_type(2))) float v2f;
typedef __attribute__((ext_vector_type(8))) float v8f;

__device__ __forceinline__ float sigmoidf_(float x) { return 1.0f / (1.0f + expf(-x)); }
__device__ __forceinline__ float softplusf_(float x) {
  return fmaxf(x, 0.0f) + log1pf(expf(-fabsf(x)));
}

// ---------------------------------------------------------------- k0: init
__global__ void k0_init(float* stats) {
  if (threadIdx.x < 768) stats[threadIdx.x] = 0.0f;
}

// ------------------------------------------------- k1: big GEMM + BN1 stats
// Persistent blocks: grid (GX, 4). Block owns 64 columns of W^T in LDS and
// loops over 32-row slabs of the virtual concat matrix [self|nbr|edge].
// 8 waves = 2 row-tiles x 4 col-tiles of 16x16; K-loop of 80 WMMA f32 16x16x4.
// b_full is omitted: a per-column constant cancels in batchnorm.
__global__ __launch_bounds__(256) void k1_gemm_stats(
    const float* __restrict__ atom_in, const float* __restrict__ nbr_fea,
    const int* __restrict__ nbr_idx, const float* __restrict__ Wfull,
    float* __restrict__ gated, float* __restrict__ gsum1, float* __restrict__ gsq1)
{
  __shared__ float sA[32 * LDA];   // 41.5 KB staged A-tile (rows x 320)
  __shared__ float sW[64 * LDW];   // 82.9 KB W^T slice: sW[c][k]
  __shared__ int   sRowN[32];
  __shared__ int   sRowJ[32];
  __shared__ float sStat[128];

  const int tid = threadIdx.x;
  const int cb0 = blockIdx.y * 64;

  // Load transposed W slice once per (persistent) block: coalesced global reads.
  for (int idx = tid; idx < 64 * KDIM; idx += 256) {
    int c = idx & 63, k = idx >> 6;
    sW[c * LDW + k] = Wfull[k * CDIM + cb0 + c];
  }

  const int wave = tid >> 5, lane = tid & 31;
  const int lg = lane >> 4, lr = lane & 15; // lane-group, lane-in-group
  const int rt = wave & 1, ct = wave >> 1;  // row-tile, col-tile of this wave
  const int cloc = ct * 16 + lr;            // local column owned by this lane

  float runS = 0.0f, runQ = 0.0f;           // running per-column sum / sumsq

  for (int slab = blockIdx.x; slab < NM / 32; slab += gridDim.x) {
    __syncthreads(); // previous iteration's LDS readers done (also W-load fence)
    if (tid < 32) {
      int r = slab * 32 + tid;
      int n = r / MNBR;
      int m = r - n * MNBR;
      sRowN[tid] = n;
      sRowJ[tid] = nbr_idx[n * MNBR + m];
    }
    __syncthreads();
    // Stage self features (gather rows are L2-resident): 32 x 128 f
    #pragma unroll
    for (int i = 0; i < 4; ++i) {
      int idx = tid + i * 256;
      int row = idx >> 5, q = idx & 31;
      float4 v = ((const float4*)(atom_in + (size_t)sRowN[row] * AF))[q];
      *(float4*)(&sA[row * LDA + q * 4]) = v;
    }
    // Stage gathered neighbor features: 32 x 128 f
    #pragma unroll
    for (int i = 0; i < 4; ++i) {
      int idx = tid + i * 256;
      int row = idx >> 5, q = idx & 31;
      float4 v = ((const float4*)(atom_in + (size_t)sRowJ[row] * AF))[q];
      *(float4*)(&sA[row * LDA + AF + q * 4]) = v;
    }
    // Stage edge features: 32 x 64 f
    #pragma unroll
    for (int i = 0; i < 2; ++i) {
      int idx = tid + i * 256;
      int row = idx >> 4, q = idx & 15;
      float4 v = ((const float4*)(nbr_fea + (size_t)(slab * 32 + row) * EF))[q];
      *(float4*)(&sA[row * LDA + 2 * AF + q * 4]) = v;
    }
    __syncthreads();

    // 16x16 fp32 WMMA accumulation over K=320 (80 x v_wmma_f32_16x16x4_f32).
    // A frag (ISA 7.12.2): lane (lg,lr) holds row lr, K = {k+2lg, k+2lg+1}.
    // B frag mirrored from W^T; C/D: lane holds column, vgpr i + 8*lg = row.
    v8f acc = {0, 0, 0, 0, 0, 0, 0, 0};
    const float* pa = &sA[(rt * 16 + lr) * LDA + 2 * lg];
    const float* pw = &sW[cloc * LDW + 2 * lg];
    #pragma unroll 8
    for (int k = 0; k < KDIM; k += 4) {
      v2f af = *(const v2f*)(pa + k);
      v2f bf = *(const v2f*)(pw + k);
      acc = __builtin_amdgcn_wmma_f32_16x16x4_f32(false, af, false, bf,
                                                  (short)0, acc, false, false);
    }

    const int c = cb0 + cloc;
    const size_t rbase = (size_t)slab * 32 + rt * 16 + lg * 8;
    #pragma unroll
    for (int i = 0; i < 8; ++i) {
      float g = acc[i];
      gated[(rbase + i) * CDIM + c] = g;
      runS += g;
      runQ += g * g;
    }
  }

  // Block-level column-stat reduction, then one global atomic per column.
  __syncthreads();
  if (tid < 128) sStat[tid] = 0.0f;
  __syncthreads();
  atomicAdd(&sStat[cloc], runS);
  atomicAdd(&sStat[64 + cloc], runQ);
  __syncthreads();
  if (tid < 64)       atomicAdd(&gsum1[cb0 + tid], sStat[tid]);
  else if (tid < 128) atomicAdd(&gsq1[cb0 + tid - 64], sStat[tid]);
}

// -------------------------- k3: BN1 + sigmoid*softplus + sum over M + stats2
__global__ __launch_bounds__(256) void k3_reduce(
    const float* __restrict__ gated, const float* __restrict__ gsum1,
    const float* __restrict__ gsq1, const float* __restrict__ gamma1,
    const float* __restrict__ beta1, float* __restrict__ nbr_sumed,
    float* __restrict__ gsum2, float* __restrict__ gsq2)
{
  __shared__ float s2s[128], s2q[128];
  const int tid = threadIdx.x;
  const int c = tid & 127, sub = tid >> 7;
  const float invNM = 1.0f / (float)NM;
  const int cc = c + 128;
  float mF = gsum1[c] * invNM;
  float vF = gsq1[c] * invNM - mF * mF;
  float sF = gamma1[c] * rsqrtf(vF + 1e-5f);
  float tF = beta1[c] - mF * sF;
  float mC = gsum1[cc] * invNM;
  float vC = gsq1[cc] * invNM - mC * mC;
  float sC = gamma1[cc] * rsqrtf(vC + 1e-5f);
  float tC = beta1[cc] - mC * sC;

  float pS = 0.0f, pQ = 0.0f;
  int n0 = blockIdx.x * 100 + sub * 50;
  for (int i = 0; i < 50; ++i) {
    int n = n0 + i;
    const float* row = gated + (size_t)n * MNBR * CDIM;
    float acc = 0.0f;
    #pragma unroll
    for (int m = 0; m < MNBR; ++m) {
      float gf = row[m * CDIM + c] * sF + tF;
      float gc = row[m * CDIM + cc] * sC + tC;
      acc += sigmoidf_(gf) * softplusf_(gc);
    }
    nbr_sumed[(size_t)n * AF + c] = acc;
    pS += acc;
    pQ += acc * acc;
  }
  if (tid < 128) { s2s[tid] = 0.0f; s2q[tid] = 0.0f; }
  __syncthreads();
  atomicAdd(&s2s[c], pS);
  atomicAdd(&s2q[c], pQ);
  __syncthreads();
  if (tid < 128) {
    atomicAdd(&gsum2[tid], s2s[tid]);
    atomicAdd(&gsq2[tid], s2q[tid]);
  }
}

// ------------------------------------------------ k4: skip GEMM (WMMA fp32)
__global__ __launch_bounds__(256) void k4_skip(
    const float* __restrict__ atom_in, const float* __restrict__ Wskip,
    const float* __restrict__ bskip, float* __restrict__ skip)
{
  __shared__ float sW2[128 * LDW2]; // W_skip^T, 67.6 KB
  __shared__ float sA2[16 * LDW2];  // 8.4 KB
  const int tid = threadIdx.x;
  for (int idx = tid; idx < 128 * 128; idx += 256) {
    int k = idx >> 7, c = idx & 127;
    sW2[c * LDW2 + k] = Wskip[idx];
  }
  const int wave = tid >> 5, lane = tid & 31;
  const int lg = lane >> 4, lr = lane & 15;
  const int c = wave * 16 + lr;
  const float bs = bskip[c];

  for (int rt = blockIdx.x; rt < NATOM / 16; rt += gridDim.x) {
    __syncthreads();
    #pragma unroll
    for (int i = 0; i < 2; ++i) {
      int idx = tid + i * 256;
      int row = idx >> 5, q = idx & 31;
      float4 v = ((const float4*)(atom_in + ((size_t)rt * 16 + row) * AF))[q];
      *(float4*)(&sA2[row * LDW2 + q * 4]) = v;
    }
    __syncthreads();
    v8f acc = {0, 0, 0, 0, 0, 0, 0, 0};
    const float* pa = &sA2[lr * LDW2 + 2 * lg];
    const float* pw = &sW2[c * LDW2 + 2 * lg];
    #pragma unroll
    for (int k = 0; k < AF; k += 4) {
      v2f af = *(const v2f*)(pa + k);
      v2f bf = *(const v2f*)(pw + k);
      acc = __builtin_amdgcn_wmma_f32_16x16x4_f32(false, af, false, bf,
                                                  (short)0, acc, false, false);
    }
    #pragma unroll
    for (int i = 0; i < 8; ++i)
      skip[((size_t)rt * 16 + lg * 8 + i) * AF + c] = acc[i] + bs;
  }
}

// ------------------------------------------------- k5: BN2 + softplus output
__global__ __launch_bounds__(256) void k5_out(
    const float* __restrict__ nbr_sumed, const float* __restrict__ skip,
    const float* __restrict__ gsum2, const float* __restrict__ gsq2,
    const float* __restrict__ gamma2, const float* __restrict__ beta2,
    float* __restrict__ out)
{
  size_t idx = (size_t)blockIdx.x * 256 + threadIdx.x;
  int c = (int)(idx & 127);
  const float invN = 1.0f / (float)NATOM;
  float m = gsum2[c] * invN;
  float v = gsq2[c] * invN - m * m;
  float s = gamma2[c] * rsqrtf(v + 1e-5f);
  float t = beta2[c] - m * s;
  float y = nbr_sumed[idx] * s + t;
  out[idx] = softplusf_(skip[idx] + y);
}

// ---------------------------------------------------------------------------
extern "C" void kernel_launch(void* const* d_in, const int* in_sizes, int n_in,
                              void* d_out, int out_size, void* d_ws, size_t ws_size,
                              hipStream_t stream)
{
  (void)in_sizes; (void)n_in; (void)out_size; (void)ws_size;
  const float* atom_in = (const float*)d_in[0];
  const float* nbr_fea = (const float*)d_in[1];
  const int*   nbr_idx = (const int*)d_in[2];
  const float* Wfull   = (const float*)d_in[3];
  // d_in[4] = b_full: unused — a per-column constant cancels in batchnorm-1.
  const float* gamma1  = (const float*)d_in[5];
  const float* beta1   = (const float*)d_in[6];
  const float* gamma2  = (const float*)d_in[7];
  const float* beta2   = (const float*)d_in[8];
  const float* Wskip   = (const float*)d_in[9];
  const float* bskip   = (const float*)d_in[10];
  float* out = (float*)d_out;

  char* ws = (char*)d_ws;
  float* gated = (float*)ws;                       // NM*256 f32 = 1.229 GB
  size_t off = (size_t)NM * CDIM * sizeof(float);
  float* nbr_sumed = (float*)(ws + off); off += (size_t)NATOM * AF * sizeof(float);
  float* skip      = (float*)(ws + off); off += (size_t)NATOM * AF * sizeof(float);
  float* stats     = (float*)(ws + off);           // 768 f32
  float* gsum1 = stats;
  float* gsq1  = stats + 256;
  float* gsum2 = stats + 512;
  float* gsq2  = stats + 640;

  hipLaunchKernelGGL(k0_init, dim3(1), dim3(768), 0, stream, stats);
  hipLaunchKernelGGL(k1_gemm_stats, dim3(512, 4), dim3(256), 0, stream,
                     atom_in, nbr_fea, nbr_idx, Wfull, gated, gsum1, gsq1);
  hipLaunchKernelGGL(k3_reduce, dim3(1000), dim3(256), 0, stream,
                     gated, gsum1, gsq1, gamma1, beta1, nbr_sumed, gsum2, gsq2);
  hipLaunchKernelGGL(k4_skip, dim3(512), dim3(256), 0, stream,
                     atom_in, Wskip, bskip, skip);
  hipLaunchKernelGGL(k5_out, dim3(50000), dim3(256), 0, stream,
                     nbr_sumed, skip, gsum2, gsq2, gamma2, beta2, out);
}